// Trittention_83975200571922
// MI455X (gfx1250) — compile-verified
//
#include <hip/hip_runtime.h>

typedef float v2f __attribute__((ext_vector_type(2)));
typedef float v8f __attribute__((ext_vector_type(8)));

#define NHEADS 8
#define DH 64
#define SEQ 128
#define DMODEL 512
#define C3 (3 * NHEADS * DH)     /* 1536 */
#define IGNORE_VAL (-1.0e6f)

// ---------------------------------------------------------------------------
// Generic fp32 WMMA GEMM: C[M x N] = A[M x K] @ W[K x N] (+ bias[N] if given)
// block = 256 threads = 8 waves; each wave owns one 16x16 output tile,
// 8 waves tile 128 columns; grid = (N/128, M/16). Uses V_WMMA_F32_16X16X4_F32.
// ---------------------------------------------------------------------------
__global__ __launch_bounds__(256)
void gemm_bias_kernel(const float* __restrict__ A, const float* __restrict__ W,
                      const float* __restrict__ bias, float* __restrict__ C,
                      int M, int N, int K)
{
    const int lane = threadIdx.x & 31;
    const int wave = threadIdx.x >> 5;
    const int n0 = blockIdx.x * 128 + wave * 16;
    const int m0 = blockIdx.y * 16;
    if (n0 >= N || m0 >= M) return;      // whole-wave uniform

    const int r  = lane & 15;            // M (for A) / N (for B,D) index
    const int hi = lane >> 4;            // half-wave select
    const int ko = hi * 2;               // K offset 0 or 2 per 16x4 layout

    v8f acc = {};
    const float* __restrict__ arow = A + (size_t)(m0 + r) * K;
    const float* __restrict__ wcol = W + (n0 + r);
#pragma unroll 4
    for (int k = 0; k < K; k += 4) {
        v2f a, b;
        a.x = arow[k + ko];
        a.y = arow[k + ko + 1];
        b.x = wcol[(size_t)(k + ko) * N];
        b.y = wcol[(size_t)(k + ko + 1) * N];
        acc = __builtin_amdgcn_wmma_f32_16x16x4_f32(
                  false, a, false, b, (short)0, acc, false, false);
    }

    const int   col   = n0 + r;
    const float bv    = bias ? bias[col] : 0.0f;
    const int   mbase = m0 + hi * 8;     // D vgpr i: lanes0-15 -> M=i, 16-31 -> M=8+i
#pragma unroll
    for (int i = 0; i < 8; ++i)
        C[(size_t)(mbase + i) * N + col] = acc[i] + bv;
}

// ---------------------------------------------------------------------------
// Fused trittention core: one workgroup per (b, head, q).
//   scores[s,t] = sum_h (k1[s,h]*q[h]) * k2[t,h]   (128x64x128 WMMA in LDS)
//   mask t>q -> IGNORE, scale 1/Dh, joint softmax over 128*128,
//   As[s] = sum_t P, At[t] = sum_s P,
//   z[h]  = (As . va[:,h] + At . vb[:,h]) / sumExp + b_v[h]
// ---------------------------------------------------------------------------
__global__ __launch_bounds__(256)
void trit_kernel(const float* __restrict__ kkq, const float* __restrict__ va,
                 const float* __restrict__ vb,  const float* __restrict__ b_v,
                 float* __restrict__ z)
{
    __shared__ float k1w[SEQ * 65];      // k1 * diag(q), padded stride 65
    __shared__ float k2s[SEQ * 65];      // k2, padded
    __shared__ float sc [SEQ * 132];     // scores / probs tile, padded stride 132
    __shared__ float wq [DH];
    __shared__ float red[8];
    __shared__ float As [SEQ];
    __shared__ float At [SEQ];
    __shared__ float zp [4][DH];
    __shared__ float gmaxs, gsums;

    const int tid  = threadIdx.x;
    const int lane = tid & 31;
    const int wave = tid >> 5;

    const int qidx = blockIdx.x & (SEQ - 1);
    const int head = (blockIdx.x >> 7) & (NHEADS - 1);
    const int b    = blockIdx.x >> 10;

    const float* __restrict__ kkq_b = kkq + (size_t)b * SEQ * C3;

    if (tid < DH)
        wq[tid] = kkq_b[(size_t)qidx * C3 + 2 * NHEADS * DH + head * DH + tid];
    __syncthreads();

    // stage k1*w and k2 tiles (8192 elems each, 32 per thread)
    for (int e = tid; e < SEQ * DH; e += 256) {
        const int s = e >> 6, h = e & 63;
        k1w[s * 65 + h] = kkq_b[(size_t)s * C3 + head * DH + h] * wq[h];
        k2s[s * 65 + h] = kkq_b[(size_t)s * C3 + NHEADS * DH + head * DH + h];
    }
    __syncthreads();

    // scores = k1w @ k2s^T via fp32 WMMA; wave w owns rows [16w,16w+16)
    {
        const int r  = lane & 15;
        const int hi = lane >> 4;
        const int ko = hi * 2;
        const int s0 = wave * 16;
        const float* __restrict__ ar = &k1w[(s0 + r) * 65];
        for (int tt = 0; tt < 8; ++tt) {
            const int t0 = tt * 16;
            const float* __restrict__ br = &k2s[(t0 + r) * 65];
            v8f acc = {};
#pragma unroll 4
            for (int k = 0; k < DH; k += 4) {
                v2f a, bfr;
                a.x   = ar[k + ko];
                a.y   = ar[k + ko + 1];
                bfr.x = br[k + ko];        // B[k][t] = k2[t][k]
                bfr.y = br[k + ko + 1];
                acc = __builtin_amdgcn_wmma_f32_16x16x4_f32(
                          false, a, false, bfr, (short)0, acc, false, false);
            }
            const int col = t0 + r;
#pragma unroll
            for (int i = 0; i < 8; ++i)
                sc[(s0 + hi * 8 + i) * 132 + col] = acc[i];
        }
    }
    __syncthreads();

    // mask (t > q), scale by 1/Dh, joint softmax over all 128*128 entries.
    // thread owns half a row: 64 contiguous t values.
    const float scale = 1.0f / (float)DH;
    const int   srow  = tid >> 1;
    const int   tbase = (tid & 1) * 64;
    float* __restrict__ row = &sc[srow * 132 + tbase];

    float lmax = -3.4e38f;
#pragma unroll 4
    for (int j = 0; j < 64; ++j) {
        const int t = tbase + j;
        float v = (t > qidx) ? IGNORE_VAL : row[j];
        v *= scale;
        row[j] = v;
        lmax = fmaxf(lmax, v);
    }
    for (int off = 16; off > 0; off >>= 1)
        lmax = fmaxf(lmax, __shfl_xor(lmax, off, 32));
    if (lane == 0) red[wave] = lmax;
    __syncthreads();
    if (tid == 0) {
        float m = red[0];
        for (int i = 1; i < 8; ++i) m = fmaxf(m, red[i]);
        gmaxs = m;
    }
    __syncthreads();
    const float gmax = gmaxs;

    float lsum = 0.0f;
#pragma unroll 4
    for (int j = 0; j < 64; ++j) {
        const float p = __expf(row[j] - gmax);
        row[j] = p;
        lsum += p;
    }
    for (int off = 16; off > 0; off >>= 1)
        lsum += __shfl_xor(lsum, off, 32);
    if (lane == 0) red[wave] = lsum;
    __syncthreads();
    if (tid == 0) {
        float s = 0.0f;
        for (int i = 0; i < 8; ++i) s += red[i];
        gsums = s;
    }
    __syncthreads();

    // marginal sums of unnormalized P: threads 0-127 rows, 128-255 columns
    if (tid < SEQ) {
        const float* __restrict__ rr = &sc[tid * 132];
        float a = 0.0f;
        for (int t = 0; t < SEQ; ++t) a += rr[t];
        As[tid] = a;
    } else {
        const int t = tid - SEQ;
        float a = 0.0f;
        for (int s = 0; s < SEQ; ++s) a += sc[s * 132 + t];
        At[t] = a;
    }
    __syncthreads();

    // z[h] = (As . va + At . vb)/sum + b_v ; 4 groups of 64 threads
    {
        const int h = tid & 63;
        const int g = tid >> 6;
        const float* __restrict__ vsrc = (g < 2) ? va : vb;
        const float* __restrict__ wsum = (g < 2) ? As : At;
        const int sbase = (g & 1) * 64;
        float acc = 0.0f;
        for (int s = 0; s < 64; ++s) {
            const int si = sbase + s;
            acc += wsum[si] * vsrc[((size_t)(b * SEQ + si) * NHEADS + head) * DH + h];
        }
        zp[g][h] = acc;
    }
    __syncthreads();
    if (tid < DH) {
        const float invS = 1.0f / gsums;
        const float zz = (zp[0][tid] + zp[1][tid] + zp[2][tid] + zp[3][tid]) * invS
                       + b_v[head * DH + tid];
        z[((size_t)(b * SEQ + qidx) * NHEADS + head) * DH + tid] = zz;
    }
}

// ---------------------------------------------------------------------------
extern "C" void kernel_launch(void* const* d_in, const int* in_sizes, int n_in,
                              void* d_out, int out_size, void* d_ws, size_t ws_size,
                              hipStream_t stream)
{
    (void)in_sizes; (void)n_in; (void)out_size; (void)ws_size;
    const float* x    = (const float*)d_in[0];
    const float* Wkkq = (const float*)d_in[1];
    const float* bkkq = (const float*)d_in[2];
    const float* Wva  = (const float*)d_in[3];
    const float* Wvb  = (const float*)d_in[4];
    const float* b_v  = (const float*)d_in[5];
    const float* Wout = (const float*)d_in[6];
    const float* bout = (const float*)d_in[7];
    float* out = (float*)d_out;

    float* ws  = (float*)d_ws;
    const int M = 2 * SEQ;                      // 256 rows (B*S)
    float* kkq = ws;                            // M * 1536
    float* va  = kkq + (size_t)M * C3;          // M * 512
    float* vb  = va  + (size_t)M * DMODEL;      // M * 512
    float* z   = vb  + (size_t)M * DMODEL;      // M * 512

    dim3 blk(256);
    // projections
    gemm_bias_kernel<<<dim3(C3 / 128, M / 16), blk, 0, stream>>>(
        x, Wkkq, bkkq, kkq, M, C3, DMODEL);
    gemm_bias_kernel<<<dim3(DMODEL / 128, M / 16), blk, 0, stream>>>(
        x, Wva, nullptr, va, M, DMODEL, DMODEL);
    gemm_bias_kernel<<<dim3(DMODEL / 128, M / 16), blk, 0, stream>>>(
        x, Wvb, nullptr, vb, M, DMODEL, DMODEL);
    // fused cubic attention: one WG per (b, head, q)
    trit_kernel<<<dim3(2 * NHEADS * SEQ), blk, 0, stream>>>(kkq, va, vb, b_v, z);
    // output projection
    gemm_bias_kernel<<<dim3(DMODEL / 128, M / 16), blk, 0, stream>>>(
        z, Wout, bout, out, M, DMODEL, DMODEL);
}